// LoopRenderer_73203422593209
// MI455X (gfx1250) — compile-verified
//
#include <hip/hip_runtime.h>

typedef __attribute__((ext_vector_type(2))) float v2f;
typedef __attribute__((ext_vector_type(8))) float v8f;

#define IMG_W   1024
#define IMG_H   1024
#define N_CORN  32
#define THICK   3.0f

// One wave = 16-pixel-wide x 4-row block. 64 strips * 256 row-groups = 16384 waves.
// Per wave: 4 rows * 2 edge-tiles * 4 matmuls = 32x v_wmma_f32_16x16x4_f32.
__global__ __launch_bounds__(256) void LoopRenderer_wmma_kernel(
    const float* __restrict__ corners, float* __restrict__ out) {
  const int lane        = threadIdx.x & 31;
  const int waveInBlock = threadIdx.x >> 5;
  const int wave        = blockIdx.x * 8 + waveInBlock;

  const int strip   = wave & 63;   // x-strip of 16 pixels
  const int rowg    = wave >> 6;   // group of 4 rows
  const int base_xi = strip << 4;
  const int base_yi = rowg << 2;

  const int  eLane = lane & 15;
  const bool hi    = (lane >= 16);

  // Pull the tiny corner table toward the WGP early (CDNA5 prefetch path).
  __builtin_prefetch(corners, 0, 1);

  // ---- Build B-matrix fragments for both 16-edge tiles (wave-invariant) ----
  // B layout (4x16, K x N): lanes 0-15 hold {K=0,K=1} for N=lane,
  //                         lanes 16-31 hold {K=2,K=3} for N=lane-16.
  v2f   bPerp[2], bMid[2], bP0[2], bP1[2];
  float thrMid[2];

#pragma unroll
  for (int t = 0; t < 2; ++t) {
    const int e  = t * 16 + eLane;
    const int en = (e + 1) & (N_CORN - 1);
    // reference swaps (row,col) -> (x,y): c.x = corners[:,1], c.y = corners[:,0]
    const float p0x = corners[2 * e + 1],  p0y = corners[2 * e + 0];
    const float p1x = corners[2 * en + 1], p1y = corners[2 * en + 0];

    const float dx  = p0x - p1x, dy = p0y - p1y;
    const float nrm = sqrtf(dx * dx + dy * dy);
    const float inv = 1.0f / fmaxf(nrm, 1e-4f);
    const float nx  = dy * inv, ny = -dx * inv;
    const float cP  = -(p0x * nx + p0y * ny);        // -(p0 . n)

    const float mx  = 0.5f * (p0x + p1x), my = 0.5f * (p0y + p1y);
    const float hl3 = 0.5f * nrm + THICK;            // half_len + thickness
    thrMid[t] = hl3 * hl3;

    // feature order K = [u, v, u^2+v^2, 1]
    bPerp[t].x = hi ? 0.0f              : nx;
    bPerp[t].y = hi ? cP                : ny;
    bMid[t].x  = hi ? 1.0f              : -2.0f * mx;
    bMid[t].y  = hi ? mx * mx + my * my : -2.0f * my;
    bP0[t].x   = hi ? 1.0f              : -2.0f * p0x;
    bP0[t].y   = hi ? p0x * p0x + p0y * p0y : -2.0f * p0y;
    bP1[t].x   = hi ? 1.0f              : -2.0f * p1x;
    bP1[t].y   = hi ? p1x * p1x + p1y * p1y : -2.0f * p1y;
  }

  // ---- 4 image rows, each: A = [u, v, u^2+v^2, 1] for 16 pixels ----
#pragma unroll
  for (int rr = 0; rr < 4; ++rr) {
    const float v = (float)(base_yi + rr);
    const float u = (float)(base_xi + eLane);
    // A layout (16x4, M x K): lanes 0-15 hold {K=0,K=1} for M=lane,
    //                         lanes 16-31 hold {K=2,K=3} for M=lane-16.
    v2f a;
    a.x = hi ? (u * u + v * v) : u;
    a.y = hi ? 1.0f            : v;

    unsigned edgebits = 0u, pointbits = 0u;   // bit m = pixel base_xi+m

#pragma unroll
    for (int t = 0; t < 2; ++t) {
      const v8f z = {};
      // D[pixel, edge]: VGPR k -> pixel M = k (lanes 0-15) / k+8 (lanes 16-31),
      //                 edge N = lane & 15 (matches per-lane thrMid).
      v8f dPerp = __builtin_amdgcn_wmma_f32_16x16x4_f32(
          false, a, false, bPerp[t], (short)0, z, false, false);
      v8f dMid  = __builtin_amdgcn_wmma_f32_16x16x4_f32(
          false, a, false, bMid[t],  (short)0, z, false, false);
      v8f dP0   = __builtin_amdgcn_wmma_f32_16x16x4_f32(
          false, a, false, bP0[t],   (short)0, z, false, false);
      v8f dP1   = __builtin_amdgcn_wmma_f32_16x16x4_f32(
          false, a, false, bP1[t],   (short)0, z, false, false);

#pragma unroll
      for (int k = 0; k < 8; ++k) {
        const bool ec = (__builtin_fabsf(dPerp[k]) <= THICK) &&
                        (dMid[k] <= thrMid[t]);
        const bool pc = (fminf(dP0[k], dP1[k]) <= THICK * THICK);
        const unsigned me = (unsigned)__ballot(ec);
        const unsigned mp = (unsigned)__ballot(pc);
        // low half-wave -> pixel k, high half-wave -> pixel k+8; OR over 16 edges
        edgebits  |= ((me & 0xffffu) ? 1u : 0u) << k;
        edgebits  |= ((me >> 16)     ? 1u : 0u) << (k + 8);
        pointbits |= ((mp & 0xffffu) ? 1u : 0u) << k;
        pointbits |= ((mp >> 16)     ? 1u : 0u) << (k + 8);
      }
    }

    // out = edge - 2*(edge & point): {0, +1, -1}
    if (lane < 16) {
      const unsigned eb = (edgebits  >> eLane) & 1u;
      const unsigned pb = (pointbits >> eLane) & 1u;
      const float val = eb ? (pb ? -1.0f : 1.0f) : 0.0f;
      out[(base_yi + rr) * IMG_W + base_xi + eLane] = val;
    }
  }
}

extern "C" void kernel_launch(void* const* d_in, const int* in_sizes, int n_in,
                              void* d_out, int out_size, void* d_ws, size_t ws_size,
                              hipStream_t stream) {
  (void)in_sizes; (void)n_in; (void)d_ws; (void)ws_size; (void)out_size;
  const float* corners = (const float*)d_in[0];   // (32, 2) float32
  float* out = (float*)d_out;                     // (1024, 1024) float32

  // 16384 waves total: 64 x-strips * 256 row-groups; 8 waves per 256-thread block.
  dim3 grid(2048), block(256);
  LoopRenderer_wmma_kernel<<<grid, block, 0, stream>>>(corners, out);
}